// PointerDecoder_34248069218414
// MI455X (gfx1250) — compile-verified
//
#include <hip/hip_runtime.h>
#include <math.h>
#include <stdint.h>

#define HDIM 128
#define PLEN 512
#define BATCH 32
#define KBIG (PLEN * HDIM)   // 65536
#define NG   (4 * HDIM)      // 512 gate columns
#define LPAD 132             // LDS row pitch (floats) to avoid bank conflicts

typedef __attribute__((ext_vector_type(2))) float v2f;
typedef __attribute__((ext_vector_type(8))) float v8f;

// D = A(16x4,f32) * B(4x16,f32) + C(16x16,f32)  -> v_wmma_f32_16x16x4_f32
__device__ __forceinline__ v8f wmma4(v2f a, v2f b, v8f c) {
  return __builtin_amdgcn_wmma_f32_16x16x4_f32(
      /*neg_a=*/false, a, /*neg_b=*/false, b,
      /*c_mod=*/(short)0, c, /*reuse_a=*/false, /*reuse_b=*/false);
}

// Async global->LDS copy, 16B per lane (GVS mode: SGPR base + u32 VGPR offset).
// lds_off = low 32 bits of generic LDS pointer (== wave-relative LDS address).
__device__ __forceinline__ void async_b128(uint32_t lds_off, uint32_t goff,
                                           const void* base) {
  asm volatile("global_load_async_to_lds_b128 %0, %1, %2 offset:0"
               :: "v"(lds_off), "v"(goff), "s"(base) : "memory");
}
// Same, but non-temporal (streamed-once data: don't pollute L2).
__device__ __forceinline__ void async_b128_nt(uint32_t lds_off, uint32_t goff,
                                              const void* base) {
  asm volatile("global_load_async_to_lds_b128 %0, %1, %2 offset:0 th:TH_LOAD_NT"
               :: "v"(lds_off), "v"(goff), "s"(base) : "memory");
}
__device__ __forceinline__ void wait_async0() {
  asm volatile("s_wait_asynccnt 0x0" ::: "memory");
}
__device__ __forceinline__ uint32_t lds_addr(const void* p) {
  return (uint32_t)(uintptr_t)p;
}

__device__ __forceinline__ float sigm(float x) { return 1.0f / (1.0f + expf(-x)); }

// ---------------------------------------------------------------------------
// fb = h_ak @ W_f.T + b_f   (32x128) ; grid 8 (n-tiles of 16), block 32
// ---------------------------------------------------------------------------
__global__ void fb_kernel(const float* __restrict__ h_ak,
                          const float* __restrict__ W_f,
                          const float* __restrict__ b_f,
                          float* __restrict__ fb) {
  const int lane = threadIdx.x & 31;
  const int n0 = blockIdx.x * 16;
  const int mr = lane & 15, ln = lane & 15;
  const int kb = (lane >> 4) << 1;
  v8f c0 = {}, c1 = {};
  for (int k0 = 0; k0 < HDIM; k0 += 4) {
    int k = k0 + kb;
    v2f a0, a1, b;
    a0.x = h_ak[mr * HDIM + k];        a0.y = h_ak[mr * HDIM + k + 1];
    a1.x = h_ak[(mr + 16) * HDIM + k]; a1.y = h_ak[(mr + 16) * HDIM + k + 1];
    b.x  = W_f[(n0 + ln) * HDIM + k];  b.y  = W_f[(n0 + ln) * HDIM + k + 1];
    c0 = wmma4(a0, b, c0);
    c1 = wmma4(a1, b, c1);
  }
  float bias = b_f[n0 + ln];
  #pragma unroll
  for (int r = 0; r < 8; ++r) {
    int m0 = (lane < 16) ? r : (r + 8);
    fb[m0 * HDIM + n0 + ln]        = c0[r] + bias;
    fb[(m0 + 16) * HDIM + n0 + ln] = c1[r] + bias;
  }
}

// ---------------------------------------------------------------------------
// scores[b][p] = sum_h tanh(H_r[p]@W_fn.T + fb)[b][h] * beta_w[h] + beta_b
// grid 512 (one per p), block 256 (8 waves; wave w owns h-tile w*16..)
// ---------------------------------------------------------------------------
__global__ void scores_kernel(const float* __restrict__ H_r,
                              const float* __restrict__ W_fn,
                              const float* __restrict__ fb,
                              const float* __restrict__ beta_w,
                              const float* __restrict__ beta_bs,
                              float* __restrict__ scores) {
  __shared__ float sc[BATCH];
  const int p = blockIdx.x;
  const int tid = threadIdx.x;
  const int wave = tid >> 5, lane = tid & 31;
  if (tid < BATCH) sc[tid] = 0.0f;
  __syncthreads();
  const int n0 = wave * 16;
  const int mr = lane & 15, ln = lane & 15;
  const int kb = (lane >> 4) << 1;
  const float* A = H_r + (size_t)p * BATCH * HDIM;  // A[m][k] = H_r[p][m][k]
  v8f c0 = {}, c1 = {};
  for (int k0 = 0; k0 < HDIM; k0 += 4) {
    int k = k0 + kb;
    v2f a0, a1, b;
    a0.x = A[mr * HDIM + k];            a0.y = A[mr * HDIM + k + 1];
    a1.x = A[(mr + 16) * HDIM + k];     a1.y = A[(mr + 16) * HDIM + k + 1];
    b.x  = W_fn[(n0 + ln) * HDIM + k];  b.y  = W_fn[(n0 + ln) * HDIM + k + 1];
    c0 = wmma4(a0, b, c0);
    c1 = wmma4(a1, b, c1);
  }
  const float bw = beta_w[n0 + ln];
  #pragma unroll
  for (int r = 0; r < 8; ++r) {
    int m0 = (lane < 16) ? r : (r + 8);
    float v0 = tanhf(c0[r] + fb[m0 * HDIM + n0 + ln]) * bw;
    float v1 = tanhf(c1[r] + fb[(m0 + 16) * HDIM + n0 + ln]) * bw;
    atomicAdd(&sc[m0], v0);        // ds_add_f32
    atomicAdd(&sc[m0 + 16], v1);
  }
  __syncthreads();
  if (tid < BATCH) scores[tid * PLEN + p] = sc[tid] + beta_bs[0];
}

// ---------------------------------------------------------------------------
// softmax + log_softmax over P per batch row. grid 32, block 512.
// ---------------------------------------------------------------------------
__global__ void softmax_kernel(const float* __restrict__ scores,
                               float* __restrict__ beta,
                               float* __restrict__ logp) {
  __shared__ float red[PLEN];
  const int b = blockIdx.x, t = threadIdx.x;
  float s = scores[b * PLEN + t];
  red[t] = s;
  __syncthreads();
  for (int off = PLEN / 2; off > 0; off >>= 1) {
    if (t < off) red[t] = fmaxf(red[t], red[t + off]);
    __syncthreads();
  }
  float mx = red[0];
  __syncthreads();
  float e = expf(s - mx);
  red[t] = e;
  __syncthreads();
  for (int off = PLEN / 2; off > 0; off >>= 1) {
    if (t < off) red[t] += red[t + off];
    __syncthreads();
  }
  float sum = red[0];
  beta[b * PLEN + t] = e / sum;
  logp[b * PLEN + t] = (s - mx) - logf(sum);
}

// ---------------------------------------------------------------------------
// gates = b_ih + b_hh + h_ak @ W_hh.T   (seeds accumulator with plain stores)
// ---------------------------------------------------------------------------
__global__ void gates_init_kernel(const float* __restrict__ h_ak,
                                  const float* __restrict__ W_hh,
                                  const float* __restrict__ b_ih,
                                  const float* __restrict__ b_hh,
                                  float* __restrict__ gates) {
  const int lane = threadIdx.x & 31;
  const int n0 = blockIdx.x * 16;
  const int mr = lane & 15, ln = lane & 15;
  const int kb = (lane >> 4) << 1;
  v8f c0 = {}, c1 = {};
  for (int k0 = 0; k0 < HDIM; k0 += 4) {
    int k = k0 + kb;
    v2f a0, a1, b;
    a0.x = h_ak[mr * HDIM + k];         a0.y = h_ak[mr * HDIM + k + 1];
    a1.x = h_ak[(mr + 16) * HDIM + k];  a1.y = h_ak[(mr + 16) * HDIM + k + 1];
    b.x  = W_hh[(n0 + ln) * HDIM + k];  b.y = W_hh[(n0 + ln) * HDIM + k + 1];
    c0 = wmma4(a0, b, c0);
    c1 = wmma4(a1, b, c1);
  }
  float bias = b_ih[n0 + ln] + b_hh[n0 + ln];
  #pragma unroll
  for (int r = 0; r < 8; ++r) {
    int m0 = (lane < 16) ? r : (r + 8);
    gates[m0 * NG + n0 + ln]        = c0[r] + bias;
    gates[(m0 + 16) * NG + n0 + ln] = c1[r] + bias;
  }
}

// ---------------------------------------------------------------------------
// gates += (beta-scaled H_r flattened, 32x65536) @ W_ih.T  tile
// grid (32 n-tiles, 16 K-chunks of 32 p's), block 256 (8 waves)
// Double-buffered async global->LDS staging (ASYNCcnt), NT hint on the
// streamed-once W_ih; beta applied to A fragments at WMMA-fetch time.
// ---------------------------------------------------------------------------
__global__ void gemm_gates_kernel(const float* __restrict__ H_r,
                                  const float* __restrict__ beta,
                                  const float* __restrict__ W_ih,
                                  float* __restrict__ gates) {
  __shared__ float xt[2][BATCH * LPAD];  // raw H_r[p] tiles (beta applied later)
  __shared__ float wt[2][16 * LPAD];     // W_ih tiles
  __shared__ float gacc[BATCH * 16];     // cross-wave partial sums
  const int tid = threadIdx.x;
  const int wave = tid >> 5, lane = tid & 31;
  const int n0 = blockIdx.x * 16;        // gate-column tile
  const int p0 = blockIdx.y * 32;        // p chunk
  const int mr = lane & 15, ln = lane & 15;
  const int kb = (lane >> 4) << 1;

  for (int i = tid; i < BATCH * 16; i += 256) gacc[i] = 0.0f;

  auto stage = [&](int buf, int p) {
    // x tile: 32x128 floats = 1024 x 16B chunks, 4 per thread (coalesced)
    #pragma unroll
    for (int c = 0; c < 4; ++c) {
      int q = tid + c * 256;                 // chunk id 0..1023
      int b = q >> 5, h4 = (q & 31) << 2;    // 32 chunks per 128-float row
      async_b128(lds_addr(&xt[buf][b * LPAD + h4]),
                 (uint32_t)((p * BATCH * HDIM + b * HDIM + h4) * 4), H_r);
    }
    // W tile: 16x128 floats = 512 x 16B chunks, 2 per thread (NT: read once)
    #pragma unroll
    for (int c = 0; c < 2; ++c) {
      int q = tid + c * 256;
      int j = q >> 5, h4 = (q & 31) << 2;
      uint64_t off = ((uint64_t)(n0 + j) * KBIG + (uint64_t)p * HDIM + h4) * 4;
      async_b128_nt(lds_addr(&wt[buf][j * LPAD + h4]), (uint32_t)off, W_ih);
    }
  };

  stage(0, p0);
  wait_async0();
  __syncthreads();

  v8f c0 = {}, c1 = {};
  for (int pp = 0; pp < 32; ++pp) {
    const int cur = pp & 1;
    const int p = p0 + pp;
    if (pp < 31) stage(cur ^ 1, p + 1);      // overlap next tile with compute
    const float s0 = beta[mr * PLEN + p];
    const float s1 = beta[(mr + 16) * PLEN + p];
    for (int s = wave; s < 32; s += 8) {     // K split across the 8 waves
      int k = s * 4 + kb;
      v2f a0, a1, b;
      a0.x = xt[cur][mr * LPAD + k] * s0;
      a0.y = xt[cur][mr * LPAD + k + 1] * s0;
      a1.x = xt[cur][(mr + 16) * LPAD + k] * s1;
      a1.y = xt[cur][(mr + 16) * LPAD + k + 1] * s1;
      b.x  = wt[cur][ln * LPAD + k];
      b.y  = wt[cur][ln * LPAD + k + 1];
      c0 = wmma4(a0, b, c0);
      c1 = wmma4(a1, b, c1);
    }
    if (pp < 31) {
      wait_async0();                         // next tile landed in LDS
      __syncthreads();                       // ...and everyone is done with cur
    }
  }
  __syncthreads();
  // reduce the 8 waves' partial C fragments in LDS
  #pragma unroll
  for (int r = 0; r < 8; ++r) {
    int m0 = (lane < 16) ? r : (r + 8);
    atomicAdd(&gacc[m0 * 16 + ln], c0[r]);
    atomicAdd(&gacc[(m0 + 16) * 16 + ln], c1[r]);
  }
  __syncthreads();
  // one global atomic per output element per block (K-split accumulation)
  for (int i = tid; i < BATCH * 16; i += 256) {
    int b = i >> 4, j = i & 15;
    atomicAdd(&gates[b * NG + n0 + j], gacc[i]);
  }
}

// ---------------------------------------------------------------------------
// LSTM cell pointwise. grid 16, block 256. out[0:4096]=h_new, [4096:8192]=c_new
// ---------------------------------------------------------------------------
__global__ void lstm_kernel(const float* __restrict__ gates,
                            const float* __restrict__ hidden,
                            float* __restrict__ out) {
  int idx = blockIdx.x * blockDim.x + threadIdx.x;
  if (idx >= BATCH * HDIM) return;
  int b = idx >> 7, h = idx & 127;
  float ig = gates[b * NG + h];
  float fg = gates[b * NG + HDIM + h];
  float gg = gates[b * NG + 2 * HDIM + h];
  float og = gates[b * NG + 3 * HDIM + h];
  float c_new = sigm(fg) * hidden[idx] + sigm(ig) * tanhf(gg);
  float h_new = sigm(og) * tanhf(c_new);
  out[idx] = h_new;
  out[BATCH * HDIM + idx] = c_new;
}

// ---------------------------------------------------------------------------
extern "C" void kernel_launch(void* const* d_in, const int* in_sizes, int n_in,
                              void* d_out, int out_size, void* d_ws, size_t ws_size,
                              hipStream_t stream) {
  const float* h_ak   = (const float*)d_in[0];
  const float* H_r    = (const float*)d_in[1];
  const float* hidden = (const float*)d_in[2];
  const float* W_f    = (const float*)d_in[3];
  const float* b_f    = (const float*)d_in[4];
  const float* W_fn   = (const float*)d_in[5];
  const float* beta_w = (const float*)d_in[6];
  const float* beta_b = (const float*)d_in[7];
  const float* W_ih   = (const float*)d_in[8];
  const float* W_hh   = (const float*)d_in[9];
  const float* b_ih   = (const float*)d_in[10];
  const float* b_hh   = (const float*)d_in[11];
  float* out = (float*)d_out;
  float* ws  = (float*)d_ws;

  float* fb     = ws;                      // 32*128   = 4096
  float* scores = ws + 4096;               // 32*512   = 16384
  float* beta   = ws + 4096 + 16384;       // 32*512   = 16384
  float* gates  = ws + 4096 + 2 * 16384;   // 32*512   = 16384  (~208 KB total)

  fb_kernel<<<8, 32, 0, stream>>>(h_ak, W_f, b_f, fb);
  scores_kernel<<<PLEN, 256, 0, stream>>>(H_r, W_fn, fb, beta_w, beta_b, scores);
  softmax_kernel<<<BATCH, PLEN, 0, stream>>>(scores, beta, out + 2 * BATCH * HDIM);
  gates_init_kernel<<<NG / 16, 32, 0, stream>>>(h_ak, W_hh, b_ih, b_hh, gates);
  gemm_gates_kernel<<<dim3(NG / 16, 16), 256, 0, stream>>>(H_r, beta, W_ih, gates);
  lstm_kernel<<<16, 256, 0, stream>>>(gates, hidden, out);
}